// FieldAwareFactorizationMachineModel_22007412425277
// MI455X (gfx1250) — compile-verified
//
#include <hip/hip_runtime.h>

typedef float v2f __attribute__((ext_vector_type(2)));
typedef float v8f __attribute__((ext_vector_type(8)));

#define NF 10
#define VFIELD 10000
#define VTOT 100000
#define ED 16
#define TABF (VTOT * ED)        // floats per field table (1,600,000); 10*TABF fits in i32

// Upper-triangle pair enumeration (i<j), 45 pairs. constexpr -> folds to immediates
// (no constant-memory traffic). Grouped in 3 chunks of 15 pairs so each chunk
// issues 30 independent gathers before any consume (MLP ~30 outstanding per wave).
static constexpr int PI_[45] = {
    0,0,0,0,0,0,0,0,0, 1,1,1,1,1,1,1,1, 2,2,2,2,2,2,2,
    3,3,3,3,3,3, 4,4,4,4,4, 5,5,5,5, 6,6,6, 7,7, 8};
static constexpr int PJ_[45] = {
    1,2,3,4,5,6,7,8,9, 2,3,4,5,6,7,8,9, 3,4,5,6,7,8,9,
    4,5,6,7,8,9, 5,6,7,8,9, 6,7,8,9, 7,8,9, 8,9, 9};

// One block = 16 samples. Thread (sl, d): sl = sample-in-block (0..15), d = dim (0..15).
// Phase 1: chunked gathers + FMA per (sample, d) -> 64B coalesced rows per half-wave.
// Phase 2: wave 0 reduces over d with chained V_WMMA_F32_16X16X4_F32 (B = ones), full f32.
__global__ __launch_bounds__(256) void ffm_kernel(
    const int*   __restrict__ x,        // (B, F)
    const int*   __restrict__ offsets,  // (F)
    const float* __restrict__ emb,      // (F, V, D)
    const float* __restrict__ lw,       // (V, 1)
    const float* __restrict__ lb,       // (1,)
    float*       __restrict__ out)      // (B,)
{
    __shared__ int   s_idx[16][NF];
    __shared__ float s_acc[16][16];

    const int t  = threadIdx.x;
    const int sl = t >> 4;              // local sample 0..15
    const int d  = t & 15;              // dim 0..15
    const int s  = blockIdx.x * 16 + sl;

    if (d < NF) {
        s_idx[sl][d] = x[s * NF + d] + offsets[d];
    }
    __syncthreads();

    // 32-bit row offsets (floats) for this thread's dim within each table row.
    int off[NF];
    #pragma unroll
    for (int f = 0; f < NF; ++f) off[f] = s_idx[sl][f] * ED + d;

    // acc(s,d) = sum_{i<j} emb[j, idx_i, d] * emb[i, idx_j, d]
    // Chunks of 15 pairs: 30 independent loads issued back-to-back, then 15 FMAs.
    float acc = 0.0f;
    #pragma unroll
    for (int g = 0; g < 45; g += 15) {
        float av[15], bv[15];
        #pragma unroll
        for (int p = 0; p < 15; ++p) {
            const int i = PI_[g + p];
            const int j = PJ_[g + p];
            av[p] = emb[j * TABF + off[i]];
            bv[p] = emb[i * TABF + off[j]];
        }
        #pragma unroll
        for (int p = 0; p < 15; ++p) acc = fmaf(av[p], bv[p], acc);
    }

    // Fold linear term into per-d partials (d<10 each take one weight); the WMMA
    // d-reduction then yields ffm + linear together.
    if (d < NF) acc += lw[s_idx[sl][d]];

    s_acc[sl][d] = acc;
    __syncthreads();

    // Phase 2: wave 0 (EXEC all-ones within the wave) reduces the 16x16 partials.
    if (t < 32) {
        const int lane = t;
        const int m = lane & 15;        // A-matrix row M = lane % 16
        const int h = lane >> 4;        // lane half selects K+{0,1} vs K+{2,3}

        v8f c = {};
        #pragma unroll
        for (int kc = 0; kc < 4; ++kc) {
            // A 16x4 f32 layout: VGPR0 = K(4kc+2h), VGPR1 = K(4kc+2h+1), row = lane%16.
            v2f a;
            a.x = s_acc[m][kc * 4 + 2 * h + 0];
            a.y = s_acc[m][kc * 4 + 2 * h + 1];
            v2f bones = {1.0f, 1.0f};   // B all-ones: C[m,n] = sum_k A[m,k], any n
            c = __builtin_amdgcn_wmma_f32_16x16x4_f32(
                    /*neg_a=*/false, a, /*neg_b=*/false, bones,
                    /*c_mod=*/(short)0, c, /*reuse_a=*/false, /*reuse_b=*/false);
        }

        // C/D layout: VGPR r, lanes 0-15 -> M=r ; lanes 16-31 -> M=8+r (columns identical).
        const float bias = lb[0];
        const int base = blockIdx.x * 16;
        if (lane == 0) {
            #pragma unroll
            for (int r = 0; r < 8; ++r) out[base + r] = c[r] + bias;
        } else if (lane == 16) {
            #pragma unroll
            for (int r = 0; r < 8; ++r) out[base + 8 + r] = c[r] + bias;
        }
    }
}

extern "C" void kernel_launch(void* const* d_in, const int* in_sizes, int n_in,
                              void* d_out, int out_size, void* d_ws, size_t ws_size,
                              hipStream_t stream) {
    (void)in_sizes; (void)n_in; (void)d_ws; (void)ws_size;
    const int*   x       = (const int*)  d_in[0];   // (16384, 10) int32
    const int*   offsets = (const int*)  d_in[1];   // (10) int32
    const float* emb     = (const float*)d_in[2];   // (10, 100000, 16) f32
    const float* lw      = (const float*)d_in[3];   // (100000, 1) f32
    const float* lb      = (const float*)d_in[4];   // (1,) f32
    float*       out     = (float*)d_out;           // (16384,) f32

    const int blocks = 16384 / 16;                  // 1024 blocks x 256 threads
    ffm_kernel<<<blocks, 256, 0, stream>>>(x, offsets, emb, lw, lb, out);
}